// ChiGAD_56255481643509
// MI455X (gfx1250) — compile-verified
//
#include <hip/hip_runtime.h>

// ---------------------------------------------------------------------------
// ChiGAD GNN pipeline for gfx1250 (MI455X), wave32 + WMMA f32 16x16x4.
//
// Algebra: feat recurrence in _poly_conv is theta-independent, so all three
// convs share P_k = L^k h (k=0..3).  h_final @ Wm1 == sum_k P_k @ Wc_k with
// Wc_k = sum_i thetas[i,k] * Wm1[i*64:(i+1)*64, :].  This cuts the dominant
// edge-scatter traffic 3x and removes the [N,192] concat buffer.
// ---------------------------------------------------------------------------

#define N_NODES   100000
#define N_EDGES   1600000
#define IN_FEATS  128
#define H_FEATS   64
#define N_CONVS   3
#define POLY_ORDER 4
#define NUM_CLASSES 2

typedef __attribute__((ext_vector_type(2))) float v2f;
typedef __attribute__((ext_vector_type(8))) float v8f;

__device__ __forceinline__ v8f wmma_f32(v2f a, v2f b, v8f c) {
  // D = A(16x4 f32) * B(4x16 f32) + C(16x16 f32)
  return __builtin_amdgcn_wmma_f32_16x16x4_f32(
      /*neg_a=*/false, a, /*neg_b=*/false, b,
      /*c_mod=*/(short)0, c, /*reuse_a=*/false, /*reuse_b=*/false);
}

// ---------------------------------------------------------------------------
__global__ __launch_bounds__(256)
void zero_f32(float* __restrict__ p, int n) {
  int stride = gridDim.x * blockDim.x;
  for (int i = blockIdx.x * blockDim.x + threadIdx.x; i < n; i += stride)
    p[i] = 0.0f;
}

__global__ __launch_bounds__(256)
void degree_kernel(const int* __restrict__ dst, float* __restrict__ deg, int E) {
  int e = blockIdx.x * blockDim.x + threadIdx.x;
  if (e < E) atomicAdd(&deg[dst[e]], 1.0f);
}

__global__ __launch_bounds__(256)
void dinv_kernel(float* __restrict__ deg, int n) {
  int i = blockIdx.x * blockDim.x + threadIdx.x;
  if (i < n) {
    float d = deg[i];
    d = d < 1.0f ? 1.0f : d;
    deg[i] = rsqrtf(d);   // in place: deg buffer becomes d^{-1/2}
  }
}

// ---------------------------------------------------------------------------
// Fused trunk: P0 = relu(relu(X@W1 + b1) @ W2 + b2).
// 1 block = 128 threads = 4 waves; 16 rows per block; wave w owns cols
// [16w,16w+16).  X tile staged in LDS; intermediate 16x64 via LDS.
// ---------------------------------------------------------------------------
__global__ __launch_bounds__(128)
void trunk_kernel(const float* __restrict__ X,
                  const float* __restrict__ W1, const float* __restrict__ b1,
                  const float* __restrict__ W2, const float* __restrict__ b2,
                  float* __restrict__ P0) {
  __shared__ float ldsX[16 * 132];   // 16 x 128 X tile (padded)
  __shared__ float ldsH[16 * 72];    // 16 x 64 relu(X@W1+b1) tile (padded)

  const int tid  = threadIdx.x;
  const int lane = tid & 31;
  const int wave = tid >> 5;
  const int mrow = lane & 15;        // M (A) or N (B/C) index within tile
  const int kh   = lane >> 4;        // K-pair selector for 16x16x4 layout
  const int row0 = blockIdx.x * 16;
  const int col0 = wave * 16;

  // cooperative X tile load (coalesced)
  for (int i = tid; i < 16 * IN_FEATS; i += 128) {
    int r = i >> 7, c = i & 127;
    ldsX[r * 132 + c] = X[(size_t)(row0 + r) * IN_FEATS + c];
  }
  __syncthreads();

  // GEMM1: K = 128 -> 32 wmma steps
  v8f acc = {};
  for (int k0 = 0; k0 < IN_FEATS; k0 += 4) {
    v2f a, b;
    a.x = ldsX[mrow * 132 + k0 + 2 * kh];
    a.y = ldsX[mrow * 132 + k0 + 2 * kh + 1];
    b.x = W1[(size_t)(k0 + 2 * kh)     * H_FEATS + col0 + mrow];
    b.y = W1[(size_t)(k0 + 2 * kh + 1) * H_FEATS + col0 + mrow];
    acc = wmma_f32(a, b, acc);
  }
  {
    float bias = b1[col0 + mrow];
    #pragma unroll
    for (int v = 0; v < 8; ++v) {
      int r = v + 8 * kh;                       // C/D row = VGPR + 8*(lane>>4)
      float val = acc[v] + bias;
      ldsH[r * 72 + col0 + mrow] = val > 0.0f ? val : 0.0f;
    }
  }
  __syncthreads();

  // GEMM2: K = 64 -> 16 wmma steps (A from LDS)
  v8f acc2 = {};
  for (int k0 = 0; k0 < H_FEATS; k0 += 4) {
    v2f a, b;
    a.x = ldsH[mrow * 72 + k0 + 2 * kh];
    a.y = ldsH[mrow * 72 + k0 + 2 * kh + 1];
    b.x = W2[(size_t)(k0 + 2 * kh)     * H_FEATS + col0 + mrow];
    b.y = W2[(size_t)(k0 + 2 * kh + 1) * H_FEATS + col0 + mrow];
    acc2 = wmma_f32(a, b, acc2);
  }
  {
    float bias = b2[col0 + mrow];
    #pragma unroll
    for (int v = 0; v < 8; ++v) {
      int r = v + 8 * kh;
      float val = acc2[v] + bias;
      P0[(size_t)(row0 + r) * H_FEATS + col0 + mrow] = val > 0.0f ? val : 0.0f;
    }
  }
}

// ---------------------------------------------------------------------------
// Edge scatter: agg[dst] += P[src] * dinv[src].  One wave per edge, each lane
// handles a float2 (32 lanes * 2 = 64 feats).  Native f32 global atomics.
// ---------------------------------------------------------------------------
__global__ __launch_bounds__(256)
void scatter_kernel(const float* __restrict__ P,
                    const int* __restrict__ src, const int* __restrict__ dst,
                    const float* __restrict__ dinv, float* __restrict__ agg,
                    int E) {
  int gid  = blockIdx.x * blockDim.x + threadIdx.x;
  int e    = gid >> 5;
  int lane = gid & 31;
  if (e >= E) return;
  int s = src[e];
  int d = dst[e];
  float di = dinv[s];
  const float2* pp = reinterpret_cast<const float2*>(P + (size_t)s * H_FEATS) + lane;
  float2 v = *pp;
  float* ap = agg + (size_t)d * H_FEATS + 2 * lane;
  atomicAdd(ap,     v.x * di);
  atomicAdd(ap + 1, v.y * di);
}

// P_k = P_{k-1} - agg * dinv, and clear agg for the next scatter pass.
__global__ __launch_bounds__(256)
void update_kernel(const float* __restrict__ Pprev, float* __restrict__ agg,
                   const float* __restrict__ dinv, float* __restrict__ Pk,
                   int n) {
  int i = blockIdx.x * blockDim.x + threadIdx.x;
  if (i >= n) return;
  int node = i >> 6;                 // / H_FEATS
  float a = agg[i];
  agg[i] = 0.0f;                     // ready for next pass
  Pk[i] = Pprev[i] - a * dinv[node];
}

// Wc[k] = sum_i thetas[i,k] * Wm1[i*64:(i+1)*64, :]
__global__ __launch_bounds__(256)
void wc_kernel(const float* __restrict__ thetas, const float* __restrict__ Wm1,
               float* __restrict__ Wc) {
  int t = blockIdx.x * blockDim.x + threadIdx.x;
  if (t >= POLY_ORDER * H_FEATS * H_FEATS) return;
  int k  = t >> 12;          // / (64*64)
  int rc = t & 4095;
  int r  = rc >> 6, c = rc & 63;
  float s = 0.0f;
  #pragma unroll
  for (int i = 0; i < N_CONVS; ++i)
    s += thetas[i * POLY_ORDER + k] * Wm1[(size_t)(i * H_FEATS + r) * H_FEATS + c];
  Wc[t] = s;
}

// ---------------------------------------------------------------------------
// Head: out = relu(sum_k P_k @ Wc_k + bm1) @ Wm2 + bm2.
// 4-term WMMA accumulation (P_k tiles staged in LDS), then the tiny 64->2
// projection from LDS.
// ---------------------------------------------------------------------------
__global__ __launch_bounds__(128)
void head_kernel(const float* __restrict__ P0, const float* __restrict__ P1,
                 const float* __restrict__ P2, const float* __restrict__ P3,
                 const float* __restrict__ Wc, const float* __restrict__ bm1,
                 const float* __restrict__ Wm2, const float* __restrict__ bm2,
                 float* __restrict__ out) {
  __shared__ float ldsA[16 * 72];
  __shared__ float ldsH[16 * 72];

  const int tid  = threadIdx.x;
  const int lane = tid & 31;
  const int wave = tid >> 5;
  const int mrow = lane & 15;
  const int kh   = lane >> 4;
  const int row0 = blockIdx.x * 16;
  const int col0 = wave * 16;

  const float* Ps[4] = {P0, P1, P2, P3};

  v8f acc = {};
  #pragma unroll
  for (int k = 0; k < POLY_ORDER; ++k) {
    const float* Pk = Ps[k];
    const float* Wk = Wc + k * H_FEATS * H_FEATS;
    // cooperative 16x64 A tile load
    for (int i = tid; i < 16 * H_FEATS; i += 128) {
      int r = i >> 6, c = i & 63;
      ldsA[r * 72 + c] = Pk[(size_t)(row0 + r) * H_FEATS + c];
    }
    __syncthreads();
    for (int k0 = 0; k0 < H_FEATS; k0 += 4) {
      v2f a, b;
      a.x = ldsA[mrow * 72 + k0 + 2 * kh];
      a.y = ldsA[mrow * 72 + k0 + 2 * kh + 1];
      b.x = Wk[(k0 + 2 * kh)     * H_FEATS + col0 + mrow];
      b.y = Wk[(k0 + 2 * kh + 1) * H_FEATS + col0 + mrow];
      acc = wmma_f32(a, b, acc);
    }
    __syncthreads();
  }

  {
    float bias = bm1[col0 + mrow];
    #pragma unroll
    for (int v = 0; v < 8; ++v) {
      int r = v + 8 * kh;
      float val = acc[v] + bias;
      ldsH[r * 72 + col0 + mrow] = val > 0.0f ? val : 0.0f;
    }
  }
  __syncthreads();

  // 64 -> 2 projection: 32 threads, (row, class) pairs
  if (tid < 32) {
    int r = tid >> 1, c = tid & 1;
    float s = bm2[c];
    #pragma unroll 8
    for (int j = 0; j < H_FEATS; ++j)
      s += ldsH[r * 72 + j] * Wm2[j * NUM_CLASSES + c];
    out[(size_t)(row0 + r) * NUM_CLASSES + c] = s;
  }
}

// ---------------------------------------------------------------------------
extern "C" void kernel_launch(void* const* d_in, const int* in_sizes, int n_in,
                              void* d_out, int out_size, void* d_ws, size_t ws_size,
                              hipStream_t stream) {
  const float* X      = (const float*)d_in[0];
  const int*   src    = (const int*)  d_in[1];
  const int*   dst    = (const int*)  d_in[2];
  const float* W1     = (const float*)d_in[3];
  const float* b1     = (const float*)d_in[4];
  const float* W2     = (const float*)d_in[5];
  const float* b2     = (const float*)d_in[6];
  const float* thetas = (const float*)d_in[7];
  const float* Wm1    = (const float*)d_in[8];
  const float* bm1    = (const float*)d_in[9];
  const float* Wm2    = (const float*)d_in[10];
  const float* bm2    = (const float*)d_in[11];
  float* out = (float*)d_out;
  float* ws  = (float*)d_ws;

  const int PN = N_NODES * H_FEATS;        // 6,400,000 floats per P_k
  float* dinv = ws;                        // [N]   degrees -> d^{-1/2}
  float* P0   = ws + 100352;               // 128-float aligned
  float* P1   = P0 + PN;
  float* P2   = P1 + PN;
  float* P3   = P2 + PN;
  float* agg  = P3 + PN;                   // [N,64] scatter accumulator
  float* Wc   = agg + PN;                  // [4,64,64] combined head weights

  // init: deg = 0, agg = 0 (ws is poisoned, not zeroed, by the harness)
  zero_f32<<<512, 256, 0, stream>>>(dinv, N_NODES);
  zero_f32<<<2048, 256, 0, stream>>>(agg, PN);

  // degrees & d^{-1/2}
  degree_kernel<<<(N_EDGES + 255) / 256, 256, 0, stream>>>(dst, dinv, N_EDGES);
  dinv_kernel<<<(N_NODES + 255) / 256, 256, 0, stream>>>(dinv, N_NODES);

  // fused trunk GEMMs -> P0
  trunk_kernel<<<N_NODES / 16, 128, 0, stream>>>(X, W1, b1, W2, b2, P0);

  // combined head weights (tiny)
  wc_kernel<<<(POLY_ORDER * H_FEATS * H_FEATS + 255) / 256, 256, 0, stream>>>(
      thetas, Wm1, Wc);

  // P_k = L^k P0, shared across all three convs (3 scatter passes, not 9)
  float* Pbuf[4] = {P0, P1, P2, P3};
  for (int k = 1; k < POLY_ORDER; ++k) {
    scatter_kernel<<<(N_EDGES * 32 + 255) / 256, 256, 0, stream>>>(
        Pbuf[k - 1], src, dst, dinv, agg, N_EDGES);
    update_kernel<<<(PN + 255) / 256, 256, 0, stream>>>(
        Pbuf[k - 1], agg, dinv, Pbuf[k], PN);
  }

  // fused MLP head -> out
  head_kernel<<<N_NODES / 16, 128, 0, stream>>>(
      P0, P1, P2, P3, Wc, bm1, Wm2, bm2, out);
}